// OrthogonalLayer1D_21320217658193
// MI455X (gfx1250) — compile-verified
//
#include <hip/hip_runtime.h>
#include <math.h>

typedef __attribute__((ext_vector_type(2))) float v2f;
typedef __attribute__((ext_vector_type(8))) float v8f;

#define NEXP   8
#define DIM    1024
#define STRIDE 1028   // padded row stride (floats): 16B-aligned, bank-conflict-free

__launch_bounds__(32)
__attribute__((amdgpu_waves_per_eu(4)))
__global__ void orthogonal_gs_wmma_kernel(const float* __restrict__ x,
                                          float* __restrict__ out) {
  // Two batches per wave: rows 0-7 = batch0 experts, rows 8-15 = batch1.
  __shared__ float lds_x[2][NEXP][STRIDE];   // 65792 B
  __shared__ float lds_L[2][NEXP][NEXP];     // Gram -> Cholesky L (in place)

  const int  lane = threadIdx.x;             // 0..31 (wave32)
  const long b0   = (long)blockIdx.x * 2;

  // ---------------- Stage X into LDS (coalesced b128) ----------------
  // Outer loops kept rolled so only 8 float4 (32 VGPRs) are in flight per
  // row: keeps total VGPRs < 256 so 4+ waves/SIMD can hide HBM latency.
  #pragma unroll 1
  for (int bb = 0; bb < 2; ++bb) {
    const float* src = x + (b0 + bb) * (NEXP * DIM);
    #pragma unroll 1
    for (int e = 0; e < NEXP; ++e) {
      const float4* s4 = (const float4*)(src + e * DIM);
      float4 v[DIM / 128];
      #pragma unroll
      for (int t = 0; t < DIM / 128; ++t) v[t] = s4[lane + 32 * t];
      #pragma unroll
      for (int t = 0; t < DIM / 128; ++t)
        *(float4*)&lds_x[bb][e][(lane + 32 * t) * 4] = v[t];
    }
  }
  __syncthreads();

  // ---------------- Gram via V_WMMA_F32_16X16X4_F32 ----------------
  // fp32 A 16x4 layout: lanes 0-15 hold row M=lane with K={0,1}; lanes 16-31
  // hold row M=lane-16 with K={2,3}. B 4x16 mirrors it, so the SAME fragment
  // fed as A and B yields C = Xblk * Xblk^T; the two 8x8 diagonal blocks of C
  // are the per-batch Gram matrices.
  const int    mrow   = lane & 15;
  const int    koff   = (lane >> 4) * 2;
  const float* rowptr = &lds_x[mrow >> 3][mrow & 7][0];

  v8f c = {};
  #pragma unroll 8
  for (int kk = 0; kk < DIM; kk += 4) {
    v2f a = *(const v2f*)&rowptr[kk + koff];          // ds_load_2addr_b64
    c = __builtin_amdgcn_wmma_f32_16x16x4_f32(false, a, false, a,
                                              (short)0, c, false, false);
  }

  // C layout: VGPR j, lanes 0-15 -> C[j][lane]; lanes 16-31 -> C[j+8][lane-16].
  {
    const int nn = lane & 15;
    const int bb = lane >> 4;
    if (nn < 8) {
      #pragma unroll
      for (int j = 0; j < NEXP; ++j) lds_L[bb][j][nn] = c[j];
    }
  }
  __syncthreads();

  // ---------------- Tiny 8x8 Cholesky, one lane per batch ----------------
  if ((lane & 15) == 0) {
    const int bb = lane >> 4;
    float (*G)[NEXP] = lds_L[bb];
    for (int i = 0; i < NEXP; ++i) {
      for (int j = 0; j <= i; ++j) {
        float s = G[i][j];
        for (int k = 0; k < j; ++k) s -= G[i][k] * G[j][k];
        G[i][j] = (i == j) ? sqrtf(s) : s / G[j][j];
      }
    }
  }
  __syncthreads();

  // ---------------- Forward substitution: Q = L^-1 * X ----------------
  #pragma unroll 1
  for (int bb = 0; bb < 2; ++bb) {
    float Lm[NEXP][NEXP];
    float invd[NEXP];
    #pragma unroll
    for (int i = 0; i < NEXP; ++i) {
      invd[i] = 1.0f / lds_L[bb][i][i];               // broadcast LDS reads
      #pragma unroll
      for (int j = 0; j < i; ++j) Lm[i][j] = lds_L[bb][i][j];
    }
    const float* xb = &lds_x[bb][0][0];
    float*       ob = out + (b0 + bb) * (NEXP * DIM);

    #pragma unroll 1
    for (int t = 0; t < DIM / 128; ++t) {
      const int d = (lane + 32 * t) * 4;              // 4 consecutive cols/lane
      float4 q[NEXP];
      #pragma unroll
      for (int i = 0; i < NEXP; ++i)
        q[i] = *(const float4*)&xb[i * STRIDE + d];   // ds_load_b128
      #pragma unroll
      for (int i = 0; i < NEXP; ++i) {
        float4 w = q[i];
        #pragma unroll
        for (int j = 0; j < i; ++j) {
          w.x -= Lm[i][j] * q[j].x;  w.y -= Lm[i][j] * q[j].y;
          w.z -= Lm[i][j] * q[j].z;  w.w -= Lm[i][j] * q[j].w;
        }
        w.x *= invd[i]; w.y *= invd[i]; w.z *= invd[i]; w.w *= invd[i];
        q[i] = w;
      }
      #pragma unroll
      for (int i = 0; i < NEXP; ++i)
        *(float4*)&ob[i * DIM + d] = q[i];            // coalesced b128 stores
    }
  }
}

extern "C" void kernel_launch(void* const* d_in, const int* in_sizes, int n_in,
                              void* d_out, int out_size, void* d_ws, size_t ws_size,
                              hipStream_t stream) {
  const float* x   = (const float*)d_in[0];
  float*       out = (float*)d_out;
  const int nbatch = in_sizes[0] / (NEXP * DIM);   // 8192
  const int npair  = nbatch / 2;                   // 2 batches per wave/block
  orthogonal_gs_wmma_kernel<<<npair, 32, 0, stream>>>(x, out);
}